// VectorQuantizer_7713761263717
// MI455X (gfx1250) — compile-verified
//
#include <hip/hip_runtime.h>

typedef __attribute__((ext_vector_type(2))) float v2f;
typedef __attribute__((ext_vector_type(8))) float v8f;

#define DIM     64
#define KEMB    512
#define HW      4096        // H*W
#define CHW     262144      // C*H*W
#define Q_OFF   1
#define P_OFF   8388609
#define E_OFF   8388610
#define QELEMS  8388608     // B*C*H*W
#define N_TOT   131072      // B*H*W

// ---------------- workspace init: ws[0]=loss accum (float), ws[1..512]=counts ----------------
__global__ void vq_init_kernel(unsigned int* __restrict__ ws) {
  int t = blockIdx.x * blockDim.x + threadIdx.x;
  if (t < 513) ws[t] = 0u;
}

// ---------------- main: distances via f32 WMMA, argmin, all big outputs ----------------
__global__ __launch_bounds__(256) void vq_main_kernel(
    const float* __restrict__ x, const float* __restrict__ emb,
    float* __restrict__ out, unsigned int* __restrict__ ws) {
  __shared__ int sIdx[8][16];

  const int lane = threadIdx.x & 31;
  const int wv   = threadIdx.x >> 5;
  const int m    = lane & 15;     // row-in-tile (A) / embedding-in-tile (B)
  const int hi   = lane >> 4;     // half-wave selector (K phase)
  const int n0   = (blockIdx.x * 8 + wv) * 16;   // 16 consecutive spatial positions, same (b,h)
  const int b    = n0 >> 12;
  const int h    = (n0 >> 6) & 63;
  const int w0   = n0 & 63;
  const int xoff = b * CHW + h * 64 + w0;        // x[b][d][h][w0+m] = x[xoff + d*HW + m]

  // ---- A fragments: 16x64 f32 tile in WMMA 16x16x4 A layout (lane<16: k%4 in {0,1}; lane>=16: {2,3})
  v2f a[16];
#pragma unroll
  for (int c = 0; c < 16; ++c) {
    int k = 4 * c + 2 * hi;
    a[c].x = x[xoff + k * HW + m];
    a[c].y = x[xoff + (k + 1) * HW + m];
  }

  float bestv[8];
  int   besti[8];
#pragma unroll
  for (int r = 0; r < 8; ++r) { bestv[r] = 3.4e38f; besti[r] = 0; }

  // ---- 32 column tiles of 16 embeddings; score = ||e||^2 - 2*(z.e)  (||z||^2 row-constant)
  for (int t = 0; t < 32; ++t) {
    const int e = t * 16 + m;
    const float* __restrict__ ep = emb + e * DIM + 2 * hi;
    float en = 0.0f;
    v2f bb[16];
#pragma unroll
    for (int c = 0; c < 16; ++c) {
      v2f bv = *reinterpret_cast<const v2f*>(ep + 4 * c);   // 8B-aligned (even element offset)
      bb[c] = bv;
      en += bv.x * bv.x + bv.y * bv.y;
    }
    v8f acc = {0.f, 0.f, 0.f, 0.f, 0.f, 0.f, 0.f, 0.f};
#pragma unroll
    for (int c = 0; c < 16; ++c)
      acc = __builtin_amdgcn_wmma_f32_16x16x4_f32(false, a[c], false, bb[c],
                                                  (short)0, acc, false, false);
    en += __shfl_xor(en, 16, 32);                 // full ||e||^2 on all lanes
#pragma unroll
    for (int r = 0; r < 8; ++r) {                 // C layout: lane<16 -> M=r; lane>=16 -> M=r+8
      float s = en - 2.0f * acc[r];
      if (s < bestv[r]) { bestv[r] = s; besti[r] = e; }
    }
  }

  // ---- argmin reduce across the 16 lanes of each half (tie-break: lowest index, like jnp.argmin)
#pragma unroll
  for (int off = 8; off >= 1; off >>= 1) {
#pragma unroll
    for (int r = 0; r < 8; ++r) {
      float ov = __shfl_xor(bestv[r], off, 32);
      int   oi = __shfl_xor(besti[r], off, 32);
      if (ov < bestv[r] || (ov == bestv[r] && oi < besti[r])) { bestv[r] = ov; besti[r] = oi; }
    }
  }

  // lane 0 holds rows 0..7, lane 16 holds rows 8..15
  if (lane == 0 || lane == 16) {
#pragma unroll
    for (int r = 0; r < 8; ++r) {
      sIdx[wv][hi * 8 + r] = besti[r];
      atomicAdd(&ws[1 + besti[r]], 1u);           // histogram for perplexity
    }
  }
  __syncthreads();

  // ---- quantized_ste (NCHW) + commitment-loss partial:  qst = z + (q - z)
  float lacc = 0.0f;
  const int em = sIdx[wv][m];
  for (int j = 0; j < 32; ++j) {
    int d    = 2 * j + hi;
    int xidx = xoff + d * HW + m;
    float q    = emb[em * DIM + d];
    float z    = x[xidx];
    float diff = q - z;
    out[Q_OFF + xidx] = z + diff;
    lacc += diff * diff;
  }
#pragma unroll
  for (int off = 16; off >= 1; off >>= 1) lacc += __shfl_xor(lacc, off, 32);
  if (lane == 0) atomicAdd(reinterpret_cast<float*>(ws), lacc);

  // ---- one-hot encodings: 16 rows x 512 cols, float2 stores (E_OFF is 8B-aligned)
  float* __restrict__ enc = out + E_OFF + n0 * KEMB;
  for (int i = lane; i < 16 * (KEMB / 2); i += 32) {
    int r  = i >> 8;              // row in tile (256 float2 per row)
    int c  = (i & 255) * 2;       // column
    int ei = sIdx[wv][r];
    float2 v;
    v.x = (ei == c)     ? 1.0f : 0.0f;
    v.y = (ei == c + 1) ? 1.0f : 0.0f;
    *reinterpret_cast<float2*>(enc + r * KEMB + c) = v;
  }
}

// ---------------- finalize: perplexity + loss scalars ----------------
__global__ void vq_fin_kernel(const unsigned int* __restrict__ ws, float* __restrict__ out) {
  __shared__ float s[512];
  int t = threadIdx.x;
  float p = (float)ws[1 + t] * (1.0f / (float)N_TOT);
  s[t] = p * logf(p + 1e-10f);
  __syncthreads();
  for (int st = 256; st > 0; st >>= 1) {
    if (t < st) s[t] += s[t + st];
    __syncthreads();
  }
  if (t == 0) {
    out[P_OFF] = expf(-s[0]);
    float lossAcc = reinterpret_cast<const float*>(ws)[0];
    out[0] = 0.25f * lossAcc * (1.0f / (float)QELEMS);
  }
}

extern "C" void kernel_launch(void* const* d_in, const int* in_sizes, int n_in,
                              void* d_out, int out_size, void* d_ws, size_t ws_size,
                              hipStream_t stream) {
  const float* x   = (const float*)d_in[0];   // [32,64,64,64] f32
  const float* emb = (const float*)d_in[1];   // [512,64] f32
  float* out = (float*)d_out;                 // loss | quantized | perplexity | encodings
  unsigned int* ws = (unsigned int*)d_ws;     // [0]=loss accum (f32 bits), [1..512]=counts

  vq_init_kernel<<<3, 256, 0, stream>>>(ws);
  vq_main_kernel<<<N_TOT / 128, 256, 0, stream>>>(x, emb, out, ws);
  vq_fin_kernel<<<1, 512, 0, stream>>>(ws, out);
}